// TopKGate_25984552141207
// MI455X (gfx1250) — compile-verified
//
#include <hip/hip_runtime.h>
#include <hip/hip_bf16.h>
#include <math.h>

typedef __attribute__((ext_vector_type(2))) float v2f;
typedef __attribute__((ext_vector_type(4))) float v4f;
typedef __attribute__((ext_vector_type(8))) float v8f;

#define EXPERTS 16
#define SPARSITY_EPS 1e-5f

// ---------------------------------------------------------------------------
// Kernel 1: gate logits via V_WMMA_F32_16X16X4_F32, then top-2 + softmax.
// 256 threads = 8 waves; each wave computes one 16-token x 16-expert tile.
// A layout (16x4 f32): lane<16 -> row M=lane, vgpr0/1 = K=0/1
//                      lane>=16 -> row M=lane-16, vgpr0/1 = K=2/3
// B layout (4x16 f32): mirrored, N = lane&15.
// C layout: vgpr j, lanes 0-15: (M=j, N=lane); lanes 16-31: (M=j+8, N=lane-16)
// ---------------------------------------------------------------------------
__global__ __launch_bounds__(256) void gate_kernel(
    const float* __restrict__ x,      // [N, D]
    const float* __restrict__ W,      // [E, D]
    const float* __restrict__ bias,   // [E]
    float* __restrict__ g,            // [N, E]  (workspace)
    float* __restrict__ s_out,        // [N, E]  (s_concat output)
    int N, int D)
{
    __shared__ float logits[8][16][EXPERTS];

    const int wave = threadIdx.x >> 5;
    const int lane = threadIdx.x & 31;
    const int half = lane >> 4;       // 0: K pair {0,1}, 1: K pair {2,3}
    const int lr   = lane & 15;
    const int n0   = (blockIdx.x * 8 + wave) * 16;

    const float* xrow = x + (size_t)(n0 + lr) * D;  // token row (A matrix row M=lr)
    const float* wrow = W + (size_t)lr * D;         // expert row (B matrix col N=lr)

    v8f c = {0.f, 0.f, 0.f, 0.f, 0.f, 0.f, 0.f, 0.f};
    for (int k = 0; k < D; k += 4) {
        v2f a = *(const v2f*)(xrow + k + 2 * half);
        v2f b = *(const v2f*)(wrow + k + 2 * half);
        // 8 args: (neg_a, A, neg_b, B, c_mod, C, reuse_a, reuse_b)
        c = __builtin_amdgcn_wmma_f32_16x16x4_f32(
            false, a, false, b, (short)0, c, false, false);
    }

    const float be = bias[lr];
#pragma unroll
    for (int j = 0; j < 8; ++j)
        logits[wave][half * 8 + j][lr] = c[j] + be;

    __syncthreads();

    // Lanes 0-15 of each wave: one token each -> top-2, softmax over the pair.
    if (half == 0) {
        const int t = lr;
        float v0 = -INFINITY; int i0 = 0;
#pragma unroll
        for (int e = 0; e < EXPERTS; ++e) {
            float v = logits[wave][t][e];
            if (v > v0) { v0 = v; i0 = e; }
        }
        float v1 = -INFINITY; int i1 = 0;
#pragma unroll
        for (int e = 0; e < EXPERTS; ++e) {
            if (e == i0) continue;
            float v = logits[wave][t][e];
            if (v > v1) { v1 = v; i1 = e; }
        }
        // softmax over {v0, v1} with v0 >= v1 (numerically stable)
        const float e1 = __expf(v1 - v0);
        const float inv = 1.f / (1.f + e1);
        const float w0 = inv;
        const float w1 = e1 * inv;

        const size_t base = (size_t)(n0 + t) * EXPERTS;
#pragma unroll
        for (int e = 0; e < EXPERTS; ++e) {
            float ge = (e == i0) ? w0 : ((e == i1) ? w1 : 0.f);
            g[base + e]     = ge;
            s_out[base + e] = (ge < SPARSITY_EPS) ? 1.f : 0.f;
        }
    }
}

// ---------------------------------------------------------------------------
// Kernel 2: soft_averages = mean(g, axis=0); hard_averages = mean(1 - s, axis=0)
// 32 blocks: block e<16 -> soft[e]; block e>=16 -> hard[e-16]. Deterministic
// tree reduction (no float atomics). Block 0 also writes the trailing 0 scalar.
// ---------------------------------------------------------------------------
__global__ __launch_bounds__(256) void avg_kernel(
    const float* __restrict__ g, const float* __restrict__ s,
    float* __restrict__ out_soft, float* __restrict__ out_hard,
    float* __restrict__ out_scalar, int N)
{
    __shared__ float red[256];
    const int  e    = blockIdx.x & 15;
    const bool hard = blockIdx.x >= 16;
    const float* src = hard ? s : g;

    float acc = 0.f;
    for (int t = threadIdx.x; t < N; t += 256) {
        float v = src[(size_t)t * EXPERTS + e];
        acc += hard ? (1.f - v) : v;
    }
    red[threadIdx.x] = acc;
    __syncthreads();
#pragma unroll
    for (int stride = 128; stride > 0; stride >>= 1) {
        if (threadIdx.x < stride) red[threadIdx.x] += red[threadIdx.x + stride];
        __syncthreads();
    }
    if (threadIdx.x == 0) {
        float mean = red[0] / (float)N;
        if (hard) out_hard[e] = mean; else out_soft[e] = mean;
        if (blockIdx.x == 0) out_scalar[0] = 0.f;
    }
}

// ---------------------------------------------------------------------------
// Kernel 3: y[n,d] = dot(f[n,d,0:16], g[n,0:16]). One block per token.
// Streams 403 MB of f with non-temporal b128 loads (f >> 192 MB L2).
// ---------------------------------------------------------------------------
__global__ __launch_bounds__(256) void combine_kernel(
    const float* __restrict__ f,   // [N, D, E]
    const float* __restrict__ g,   // [N, E]
    float* __restrict__ y,         // [N, D]
    int D)
{
    __shared__ float gl[EXPERTS];
    const int n = blockIdx.x;
    if (threadIdx.x < EXPERTS) gl[threadIdx.x] = g[(size_t)n * EXPERTS + threadIdx.x];
    __syncthreads();

    float w[EXPERTS];
#pragma unroll
    for (int e = 0; e < EXPERTS; ++e) w[e] = gl[e];

    for (int d = threadIdx.x; d < D; d += 256) {
        const v4f* p = (const v4f*)(f + ((size_t)n * D + d) * EXPERTS);
        v4f f0 = __builtin_nontemporal_load(p + 0);
        v4f f1 = __builtin_nontemporal_load(p + 1);
        v4f f2 = __builtin_nontemporal_load(p + 2);
        v4f f3 = __builtin_nontemporal_load(p + 3);
        float acc =
            f0.x * w[0]  + f0.y * w[1]  + f0.z * w[2]  + f0.w * w[3]  +
            f1.x * w[4]  + f1.y * w[5]  + f1.z * w[6]  + f1.w * w[7]  +
            f2.x * w[8]  + f2.y * w[9]  + f2.z * w[10] + f2.w * w[11] +
            f3.x * w[12] + f3.y * w[13] + f3.z * w[14] + f3.w * w[15];
        y[(size_t)n * D + d] = acc;
    }
}

// ---------------------------------------------------------------------------
extern "C" void kernel_launch(void* const* d_in, const int* in_sizes, int n_in,
                              void* d_out, int out_size, void* d_ws, size_t ws_size,
                              hipStream_t stream) {
    const float* f = (const float*)d_in[0];  // [N, D, E]
    const float* x = (const float*)d_in[1];  // [N, D]
    const float* W = (const float*)d_in[2];  // [E, D]
    const float* b = (const float*)d_in[3];  // [E]

    const int E = in_sizes[3];            // 16
    const int D = in_sizes[2] / E;        // 768
    const int N = in_sizes[1] / D;        // 8192

    float* out      = (float*)d_out;
    float* y        = out;                       // [N*D]
    float* soft     = y + (size_t)N * D;         // [E]
    float* hard     = soft + E;                  // [E]
    float* s_concat = hard + E;                  // [N*E]
    float* scalar0  = s_concat + (size_t)N * E;  // [1]

    float* g = (float*)d_ws;                     // [N*E] workspace

    // 1) gate: 8 waves/block, 16 tokens/wave -> 128 tokens/block
    gate_kernel<<<N / 128, 256, 0, stream>>>(x, W, b, g, s_concat, N, D);
    // 2) column means + trailing scalar
    avg_kernel<<<32, 256, 0, stream>>>(g, s_concat, soft, hard, scalar0, N);
    // 3) combine (HBM-roofline pass over f)
    combine_kernel<<<N, 256, 0, stream>>>(f, g, y, D);
}